// FermiNet_65996467470370
// MI455X (gfx1250) — compile-verified
//
#include <hip/hip_runtime.h>
#include <hip/hip_bf16.h>
#include <math.h>

// FermiNet forward on MI455X (gfx1250, wave32).
// Sizes fixed by the reference: N=32, I=32, L=16, K=16, H1=128, H2=4, F_LEN=392, n_up=16.
//
// Stage 1: preprocess (eN, rN, sh0, dh0).
// Stage 2: 16x per-layer kernels (batched per-electron matvec, bandwidth bound on v).
// Stage 3: final kernel — fp32 WMMA (V_WMMA_F32_16X16X4_F32) GEMMs for dots,
//          envelope, per-wave 16x16 LU determinants, weighted sum.

typedef __attribute__((ext_vector_type(2))) float v2f;
typedef __attribute__((ext_vector_type(8))) float v8f;

#define NE    32
#define NI    32
#define NLAY  16
#define NDET  16
#define H1    128
#define NUP   16

// ---------------------------------------------------------------- stage 1
__global__ void fn_prep(const float* __restrict__ ep, const float* __restrict__ nuc,
                        float* __restrict__ sh0, float* __restrict__ dh0,
                        float* __restrict__ rN) {
    int t = threadIdx.x;                       // 256 threads
    for (int idx = t; idx < NE * NI; idx += 256) {
        int n = idx >> 5, m = idx & 31;
        float dx = ep[n * 3 + 0] - nuc[m * 3 + 0];
        float dy = ep[n * 3 + 1] - nuc[m * 3 + 1];
        float dz = ep[n * 3 + 2] - nuc[m * 3 + 2];
        float r  = sqrtf(dx * dx + dy * dy + dz * dz);
        sh0[n * H1 + 3 * m + 0] = dx;
        sh0[n * H1 + 3 * m + 1] = dy;
        sh0[n * H1 + 3 * m + 2] = dz;
        sh0[n * H1 + 96 + m]    = r;
        rN[n * NI + m]          = r;
    }
    for (int idx = t; idx < NE * NE; idx += 256) {
        int i = idx >> 5, j = idx & 31;
        float dx = ep[i * 3 + 0] - ep[j * 3 + 0];
        float dy = ep[i * 3 + 1] - ep[j * 3 + 1];
        float dz = ep[i * 3 + 2] - ep[j * 3 + 2];
        float r  = sqrtf(dx * dx + dy * dy + dz * dz);
        float* p = dh0 + idx * 4;
        p[0] = dx; p[1] = dy; p[2] = dz; p[3] = r;
    }
}

// ---------------------------------------------------------------- stage 2
// One block per electron n; 128 threads, thread o = output channel.
__global__ void __launch_bounds__(128)
fn_layer(const float* __restrict__ v, const float* __restrict__ b,
         const float* __restrict__ w, const float* __restrict__ cc,
         const float* __restrict__ sh_in, float* __restrict__ sh_out,
         const float* __restrict__ dh_in, float* __restrict__ dh_out,
         int l) {
    __shared__ float g_up[H1], g_dn[H1], sh_row[H1], dg[8];
    const int n = blockIdx.x, o = threadIdx.x;

    float su = 0.f, sd = 0.f;
    for (int j = 0;   j < NUP; ++j) su += sh_in[j * H1 + o];
    for (int j = NUP; j < NE;  ++j) sd += sh_in[j * H1 + o];
    g_up[o] = su * (1.f / 16.f);
    g_dn[o] = sd * (1.f / 16.f);
    sh_row[o] = sh_in[n * H1 + o];
    if (o < 8) {                                  // dg_up (o<4), dg_dn (o>=4)
        int d = o & 3;
        int j0 = (o < 4) ? 0 : NUP, j1 = (o < 4) ? NUP : NE;
        float s = 0.f;
        for (int j = j0; j < j1; ++j) s += dh_in[(n * NE + j) * 4 + d];
        dg[o] = s * (1.f / 16.f);
    }
    __syncthreads();

    const float* vb = v + (size_t)(l * NE + n) * 392 * H1;
    float acc = b[(l * NE + n) * H1 + o];
    for (int j = 0; j < H1; ++j) acc += sh_row[j] * vb[(      j) * H1 + o];
    for (int j = 0; j < H1; ++j) acc += g_up[j]   * vb[(128 + j) * H1 + o];
    for (int j = 0; j < H1; ++j) acc += g_dn[j]   * vb[(256 + j) * H1 + o];
    for (int d = 0; d < 4;  ++d) acc += dg[d]     * vb[(384 + d) * H1 + o];
    for (int d = 0; d < 4;  ++d) acc += dg[4 + d] * vb[(388 + d) * H1 + o];
    sh_out[n * H1 + o] = tanhf(acc) + sh_row[o];

    if (l < NLAY - 1) {                           // double stream
        int j = o >> 2, d = o & 3;
        const float* wb = w + (size_t)((l * NE + n) * NE + j) * 16;
        const float* dr = dh_in + (n * NE + j) * 4;
        float a = cc[((l * NE + n) * NE + j) * 4 + d];
        a += dr[0] * wb[0 * 4 + d] + dr[1] * wb[1 * 4 + d] +
             dr[2] * wb[2 * 4 + d] + dr[3] * wb[3 * 4 + d];
        dh_out[(n * NE + j) * 4 + d] = tanhf(a) + dr[d];
    }
}

// ---------------------------------------------------------------- stage 3
// One block, 512 threads = 16 waves; wave k handles determinant k.
__global__ void __launch_bounds__(512)
fn_final(const float* __restrict__ sh, const float* __restrict__ rN,
         const float* __restrict__ fw, const float* __restrict__ fb,
         const float* __restrict__ pi, const float* __restrict__ sigma,
         const float* __restrict__ omega, float* __restrict__ out) {
    __shared__ float phi[NDET * 2 * 16 * 16];     // 32 KB
    __shared__ float terms[NDET];

    const int tid  = threadIdx.x;
    const int k    = tid >> 5;                    // wave id == determinant
    const int lane = tid & 31;
    const int m    = lane & 15;                   // row / col within 16-group
    const int hi   = lane >> 4;                   // K-pair selector for WMMA A/B

    for (int s = 0; s < 2; ++s) {                 // spin-up / spin-dn tile
        const int i0 = s * 16, j0 = s * 16;
        v8f acc = {};
        for (int kk = 0; kk < 32; ++kk) {         // K = 128 in steps of 4
            const int kp = kk * 4 + hi * 2;
            // A: 16x4 f32, lanes 0-15 -> K={kp,kp+1}=={0,1}, lanes 16-31 -> {2,3}
            v2f a;  a.x  = fw[(k * NE + i0 + m) * H1 + kp];
                    a.y  = fw[(k * NE + i0 + m) * H1 + kp + 1];
            // B: 4x16 f32, lane n=m holds B[kp][n], B[kp+1][n]
            v2f bb; bb.x = sh[(j0 + m) * H1 + kp];
                    bb.y = sh[(j0 + m) * H1 + kp + 1];
            acc = __builtin_amdgcn_wmma_f32_16x16x4_f32(
                false, a, false, bb, (short)0, acc, false, false);
        }
        // D layout: VGPR vv, lanes 0-15 -> (M=vv, N=lane), lanes 16-31 -> (M=vv+8)
        for (int vv = 0; vv < 8; ++vv) {
            const int i = i0 + (hi ? vv + 8 : vv);
            const int j = j0 + m;
            float val = acc[vv] + fb[k * NE + i];
            float env = 0.f;
            for (int mm = 0; mm < NI; ++mm)
                env += pi[(k * NE + i) * NI + mm] *
                       __expf(-fabsf(sigma[(k * NE + i) * NI + mm]) * rN[j * NI + mm]);
            phi[((k * 2 + s) * 16 + (i - i0)) * 16 + (j - j0)] = val * env;
        }
    }
    __syncthreads();

    // LU (no pivoting): lanes 0-15 -> up matrix, 16-31 -> dn matrix; lane owns row m.
    float row[16];
    #pragma unroll
    for (int j = 0; j < 16; ++j)
        row[j] = phi[((k * 2 + hi) * 16 + m) * 16 + j];

    float det = 1.f;
    #pragma unroll
    for (int p = 0; p < 15; ++p) {
        const float prp = __shfl(row[p], p, 16);  // pivot diag (bcast in 16-group)
        det *= prp;
        const float f = row[p] / prp;             // only meaningful for m > p
        #pragma unroll
        for (int j = p + 1; j < 16; ++j) {
            const float pv = __shfl(row[j], p, 16);
            if (m > p) row[j] -= f * pv;
        }
    }
    det *= __shfl(row[15], 15, 16);               // last diagonal element

    const float du = __shfl(det, 0, 32);
    const float dd = __shfl(det, 16, 32);
    if (lane == 0) terms[k] = omega[k] * du * dd;
    __syncthreads();
    if (tid == 0) {
        float s = 0.f;
        for (int q = 0; q < NDET; ++q) s += terms[q];
        out[0] = s;
    }
}

// ---------------------------------------------------------------- launcher
extern "C" void kernel_launch(void* const* d_in, const int* in_sizes, int n_in,
                              void* d_out, int out_size, void* d_ws, size_t ws_size,
                              hipStream_t stream) {
    (void)in_sizes; (void)n_in; (void)out_size; (void)ws_size;
    const float* ep  = (const float*)d_in[0];
    const float* nuc = (const float*)d_in[1];
    const float* v   = (const float*)d_in[2];
    const float* b   = (const float*)d_in[3];
    const float* w   = (const float*)d_in[4];
    const float* c   = (const float*)d_in[5];
    const float* fw  = (const float*)d_in[6];
    const float* fb  = (const float*)d_in[7];
    const float* pi  = (const float*)d_in[8];
    const float* sg  = (const float*)d_in[9];
    const float* om  = (const float*)d_in[10];

    float* ws  = (float*)d_ws;
    float* shA = ws;                // [32][128]
    float* shB = ws + 4096;
    float* dhA = ws + 8192;         // [32][32][4]
    float* dhB = ws + 12288;
    float* rN  = ws + 16384;        // [32][32]

    fn_prep<<<1, 256, 0, stream>>>(ep, nuc, shA, dhA, rN);

    float* si = shA; float* so = shB;
    float* di = dhA; float* dn = dhB;
    for (int l = 0; l < NLAY; ++l) {
        fn_layer<<<NE, H1, 0, stream>>>(v, b, w, c, si, so, di, dn, l);
        float* t = si; si = so; so = t;
        if (l < NLAY - 1) { t = di; di = dn; dn = t; }
    }

    fn_final<<<1, 512, 0, stream>>>(si, rN, fw, fb, pi, sg, om, (float*)d_out);
}